// vMF_4827543240807
// MI455X (gfx1250) — compile-verified
//
#include <hip/hip_runtime.h>
#include <hip/hip_bf16.h>
#include <math.h>
#include <cstdint>

typedef __attribute__((ext_vector_type(2))) float v2f;
typedef __attribute__((ext_vector_type(8))) float v8f;

#define CHUNK 64
#define PAD 4
#define RSTRIDE (CHUNK + PAD)          // 68 dwords/row -> conflict-free ds_load_b64
#define WAVES_PER_BLOCK 4
#define BLOCK_THREADS (WAVES_PER_BLOCK * 32)
#define TILE_ROWS 16
#define BUF_FLOATS (TILE_ROWS * RSTRIDE)   // 1088 floats = 4352 B per buffer

// ---------------------------------------------------------------------------
// Kernel 1: scalar prep. Computes logC (64-term log I_s(kappa) series) and
// writes smu[j] = kappa * mu[j]/||mu|| into workspace, so the streaming
// kernel is a pure GEMV + scalar add.
// ---------------------------------------------------------------------------
__global__ void vmf_prep_kernel(const float* __restrict__ mu_unnorm,
                                const float* __restrict__ logkappa,
                                float* __restrict__ wsf, int D) {
  __shared__ float red[1024];
  const int tid = threadIdx.x;
  const float v = (tid < D) ? mu_unnorm[tid] : 0.0f;
  red[tid] = v * v;
  __syncthreads();
  for (int off = blockDim.x >> 1; off > 0; off >>= 1) {
    if (tid < off) red[tid] += red[tid + off];
    __syncthreads();
  }
  const float norm  = fmaxf(sqrtf(red[0]), 1e-12f);    // jnp.clip(norm, EPS)
  const float kappa = expf(logkappa[0]) + 1e-6f;        // exp(logkappa) + REG

  if (tid == 0) {
    const float s   = 0.5f * (float)D - 1.0f;           // d/2 - 1
    const float lhk = logf(0.5f * kappa);               // log(kappa/2)
    float mx = -INFINITY;
    for (int m = 0; m < 64; ++m) {
      const float fm = (float)m;
      const float lt = 2.0f * fm * lhk - lgammaf(fm + 1.0f) - lgammaf(fm + s + 1.0f);
      mx = fmaxf(mx, lt);
    }
    float sum = 0.0f;
    for (int m = 0; m < 64; ++m) {
      const float fm = (float)m;
      const float lt = 2.0f * fm * lhk - lgammaf(fm + 1.0f) - lgammaf(fm + s + 1.0f);
      sum += expf(lt - mx);
    }
    const float logI = s * lhk + mx + logf(sum);        // log I_s(kappa)
    const float NHLOG2PI = -0.91893853320467274178f;    // -0.5*log(2*pi)
    wsf[0] = (float)D * NHLOG2PI + s * logf(kappa) - logI;   // logC
  }
  if (tid < D) wsf[16 + tid] = (kappa / norm) * v;      // smu
}

// ---------------------------------------------------------------------------
// Kernel 2: bandwidth-bound streaming GEMV via V_WMMA_F32_16X16X4_F32 with
// double-buffered GLOBAL_LOAD_ASYNC_TO_LDS_B128 staging (ASYNCcnt pipelined).
// One wave per 16-row tile; B = broadcast smu chunk, preloaded to registers.
// ---------------------------------------------------------------------------
__global__ __launch_bounds__(BLOCK_THREADS) void vmf_wmma_kernel(
    const float* __restrict__ x, const float* __restrict__ wsf,
    float* __restrict__ out, int N, int D) {
  __shared__ __align__(16) float lds[WAVES_PER_BLOCK * 2 * BUF_FLOATS];  // 34,816 B

  const int lane    = threadIdx.x & 31;
  const int wave    = threadIdx.x >> 5;
  const int tile    = blockIdx.x * WAVES_PER_BLOCK + wave;
  const int rowBase = tile * TILE_ROWS;
  if (rowBase >= N) return;   // safe: no block-wide barriers below

  const float logC = wsf[0];
  const float* __restrict__ smu = wsf + 16;

  const int l16  = lane & 15;
  const int hi   = lane >> 4;     // 0: lanes 0-15, 1: lanes 16-31
  const int koff = hi * 2;        // A/B layout: VGPR0 = K0 (lo half) / K2 (hi half)

  // LDS byte offset of this wave's double buffer (flat LDS ptr low 32 bits
  // == LDS byte offset per the aperture address mapping).
  const uint32_t ldsBase =
      (uint32_t)(uintptr_t)&lds[wave * 2 * BUF_FLOATS];

  // Issue one chunk (16 rows x 64 floats) of async global->LDS B128 copies:
  // 8 instructions, 2 rows per instruction, fully coalesced, 16B aligned.
  auto issue_chunk = [&](int bufIdx, int k0) {
#pragma unroll
    for (int i = 0; i < TILE_ROWS / 2; ++i) {
      const int row = i * 2 + hi;
      const int voff = (((rowBase + row) * D) + k0 + l16 * 4) * 4;  // bytes
      const uint32_t ldsAddr =
          ldsBase + (uint32_t)((bufIdx * BUF_FLOATS + row * RSTRIDE + l16 * 4) * 4);
      asm volatile("global_load_async_to_lds_b128 %0, %1, %2"
                   :: "v"(ldsAddr), "v"(voff), "s"(x)
                   : "memory");
    }
  };

  v8f acc = {};
  const int nCh = D / CHUNK;

  issue_chunk(0, 0);   // prologue: chunk 0 -> buffer 0

  for (int t = 0; t < nCh; ++t) {
    const int k0  = t * CHUNK;
    const int cur = t & 1;

    // Pipeline: issue chunk t+1 into the other buffer, prefetch chunk t+2.
    if (t + 1 < nCh) {
      issue_chunk(1 - cur, k0 + CHUNK);
      if (k0 + 2 * CHUNK < D)
        __builtin_prefetch(x + (size_t)(rowBase + l16) * D + k0 + 2 * CHUNK + hi * 32, 0, 3);
    }

    // Preload this chunk's B operands (broadcast smu) into registers so the
    // WMMA loop has no global waits.
    v2f breg[CHUNK / 4];
#pragma unroll
    for (int j = 0; j < CHUNK / 4; ++j)
      breg[j] = *(const v2f*)&smu[k0 + j * 4 + koff];

    // Async loads complete in order: <=8 outstanding means chunk t landed.
    if (t + 1 < nCh) asm volatile("s_wait_asynccnt 0x8" ::: "memory");
    else             asm volatile("s_wait_asynccnt 0x0" ::: "memory");

    const float* wbuf = &lds[(wave * 2 + cur) * BUF_FLOATS];
#pragma unroll
    for (int j = 0; j < CHUNK / 4; ++j) {
      v2f a = *(const v2f*)&wbuf[l16 * RSTRIDE + j * 4 + koff];
      acc = __builtin_amdgcn_wmma_f32_16x16x4_f32(
          /*neg_a=*/false, a, /*neg_b=*/false, breg[j],
          /*c_mod=*/(short)0, acc, /*reuse_a=*/false, /*reuse_b=*/false);
    }
  }

  // C/D layout: VGPR v -> M=v (lanes 0-15) / M=8+v (lanes 16-31); all N equal.
  if (l16 == 0) {
    float* o = out + rowBase + hi * 8;
    float4 o0 = make_float4(acc[0] + logC, acc[1] + logC, acc[2] + logC, acc[3] + logC);
    float4 o1 = make_float4(acc[4] + logC, acc[5] + logC, acc[6] + logC, acc[7] + logC);
    *(float4*)(o + 0) = o0;
    *(float4*)(o + 4) = o1;
  }
}

// ---------------------------------------------------------------------------
extern "C" void kernel_launch(void* const* d_in, const int* in_sizes, int n_in,
                              void* d_out, int out_size, void* d_ws, size_t ws_size,
                              hipStream_t stream) {
  const float* x  = (const float*)d_in[0];
  const float* mu = (const float*)d_in[1];
  const float* lk = (const float*)d_in[2];
  float* out = (float*)d_out;
  float* wsf = (float*)d_ws;   // [0]=logC, [16..16+D) = kappa*mu/||mu||

  const int D = in_sizes[1];
  const int N = in_sizes[0] / D;

  vmf_prep_kernel<<<1, D, 0, stream>>>(mu, lk, wsf, D);

  const int tiles  = (N + TILE_ROWS - 1) / TILE_ROWS;
  const int blocks = (tiles + WAVES_PER_BLOCK - 1) / WAVES_PER_BLOCK;
  vmf_wmma_kernel<<<blocks, BLOCK_THREADS, 0, stream>>>(x, wsf, out, N, D);
}